// NCNModuleTriton_23433341567538
// MI455X (gfx1250) — compile-verified
//
#include <hip/hip_runtime.h>
#include <stdint.h>

// Problem constants (fixed by the reference): B=2, S=2048, E=1024, H=16, D=64,
// GROUP=32, num_groups=64, MODULE_L=1 -> butterfly stride 1.
// The recurrence is independent per head, so a block owns all 32 rows of a
// 2-head (128-col) window: per-thread state stays small -> no spills.
#define BB 2
#define SS 2048
#define EE 1024
#define GG 32
#define NG 64
#define CPB 128                   // columns per block (2 heads)
#define NHP 8                     // 128-col windows per row (EE / CPB)

#define STAGE_F (GG * CPB)        // 4096 floats per staged tensor
#define XJ_OFF  (2 * STAGE_F)     // float offset of ping-pong xj buffers
#define XJ_STRIDE 132             // 132 floats per buffer (swizzled, 16B-aligned)

// CDNA5 async global->LDS path (ASYNCcnt); confirmed present on this toolchain.
#if defined(__HIP_DEVICE_COMPILE__) && defined(__gfx1250__) && \
    __has_builtin(__builtin_amdgcn_global_load_async_to_lds_b128) && \
    __has_builtin(__builtin_amdgcn_s_wait_asynccnt)
#define USE_ASYNC_LDS 1
#else
#define USE_ASYNC_LDS 0
#endif

#if USE_ASYNC_LDS
// Signature learned from the round-1 diagnostic: generic pointers to 16-byte int vectors.
typedef int v4i __attribute__((vector_size(4 * sizeof(int))));
#define ASYNC_COPY_B128(gp, lp)                                                   \
    __builtin_amdgcn_global_load_async_to_lds_b128((v4i*)(gp), (v4i*)(lp), 0, 0)
#endif

__global__ __launch_bounds__(256)
void ncn_fused(const float* __restrict__ x, const float* __restrict__ xa,
               const float* __restrict__ W, float* __restrict__ out)
{
    extern __shared__ float smem[];   // [x stage 4096 | xa stage 4096 | xj ping-pong 2*132]

    const int bid = (int)blockIdx.x;       // 0..1023
    const int hp  = bid & (NHP - 1);       // which 128-col window
    const int g   = (bid >> 3) & (NG - 1); // group
    const int b   = bid >> 9;              // batch

    const int t    = (int)threadIdx.x;     // 256 threads = 8 waves
    const int r    = t >> 3;               // row in group, 0..31 (4 rows per wave)
    const int c16  = t & 7;                // 16-col chunk within the window
    // lane = (r&3)*8 + c16 : the 4 d-quarters of one (r,head) differ in lane bits 0..1

    const int colg = hp * CPB + c16 * 16;  // global column of this thread's slice
    // butterfly gather: source row for slot (g, r), stride 1
    const int srow = ((g + r) & (NG - 1)) * GG + r;
    const size_t rowwin = ((size_t)b * SS + srow) * (size_t)EE + (size_t)(hp * CPB);
    const size_t gbase  = rowwin + (size_t)(c16 * 16);

    // Swizzled xj chunk base: bank bases {0,16,32,48,4,20,36,52} -> conflict-free b128
    const int xjb = (c16 << 4) + (c16 & 4);

    float xi[16], xg[16], wi[16], wj[16];

    // Per-thread weight slices: Wi = W[0:E], Wj = W[E:2E]
    #pragma unroll
    for (int k = 0; k < 16; k += 4) {
        float4 a = *(const float4*)(W + colg + k);
        float4 c = *(const float4*)(W + EE + colg + k);
        wi[k+0]=a.x; wi[k+1]=a.y; wi[k+2]=a.z; wi[k+3]=a.w;
        wj[k+0]=c.x; wj[k+1]=c.y; wj[k+2]=c.z; wj[k+3]=c.w;
    }

#if USE_ASYNC_LDS
    {
        // One async burst stages both tensors (wave stages only its own 4 rows,
        // lane-consecutive 16B chunks -> fully coalesced). A single
        // s_wait_asynccnt orders the same-wave slice reads; no barriers needed.
        float* xrow  = smem + r * CPB;
        float* xarow = smem + STAGE_F + r * CPB;
        #pragma unroll
        for (int it = 0; it < 4; ++it) {
            const int off = c16 * 4 + it * 32;     // floats within 128-col window
            ASYNC_COPY_B128(x  + rowwin + off, xrow  + off);
            ASYNC_COPY_B128(xa + rowwin + off, xarow + off);
        }
        __builtin_amdgcn_s_wait_asynccnt(0);
        #pragma unroll
        for (int k = 0; k < 16; k += 4) {
            float4 v  = *(const float4*)(xrow  + c16 * 16 + k);
            float4 v2 = *(const float4*)(xarow + c16 * 16 + k);
            xi[k+0]=v.x;  xi[k+1]=v.y;  xi[k+2]=v.z;  xi[k+3]=v.w;
            xg[k+0]=v2.x; xg[k+1]=v2.y; xg[k+2]=v2.z; xg[k+3]=v2.w;
        }
    }
#else
    #pragma unroll
    for (int k = 0; k < 16; k += 4) {
        float4 v  = *(const float4*)(x  + gbase + k);
        float4 v2 = *(const float4*)(xa + gbase + k);
        xi[k+0]=v.x;  xi[k+1]=v.y;  xi[k+2]=v.z;  xi[k+3]=v.w;
        xg[k+0]=v2.x; xg[k+1]=v2.y; xg[k+2]=v2.z; xg[k+3]=v2.w;
    }
#endif

    // Pre-loop: row 0 publishes its initial slice into ping buffer 0.
    if (r == 0) {
        float* buf0 = smem + XJ_OFF + xjb;
        #pragma unroll
        for (int k = 0; k < 16; k += 4)
            *(float4*)(buf0 + k) = make_float4(xi[k], xi[k+1], xi[k+2], xi[k+3]);
    }

    // 32-step sequential recurrence; carried state stays in VGPRs.
    // Ping-pong xj buffers -> exactly one barrier per step: row j+1 writes the
    // OTHER buffer after its own step-j update, so it cannot race step-j readers.
    for (int j = 0; j < GG; ++j) {
        __syncthreads();

        const float* buf = smem + XJ_OFF + ((j & 1) ? XJ_STRIDE : 0) + xjb;
        float xjv[16];
        #pragma unroll
        for (int k = 0; k < 16; k += 4) {
            float4 v = *(const float4*)(buf + k);
            xjv[k+0]=v.x; xjv[k+1]=v.y; xjv[k+2]=v.z; xjv[k+3]=v.w;
        }

        // sim partial over this thread's 16 d's; 4-lane butterfly completes the
        // 64-d reduction for (r, head) entirely inside the wave32.
        float p = 0.0f;
        #pragma unroll
        for (int k = 0; k < 16; ++k) {
            p = __builtin_fmaf(xi[k], wi[k], p);
            p = __builtin_fmaf(xjv[k], wj[k], p);
        }
        p += __shfl_xor(p, 1, 32);
        p += __shfl_xor(p, 2, 32);
        const float sim = p;

        #pragma unroll
        for (int k = 0; k < 16; ++k) {
            float T  = 0.5f * xi[k] + 0.5f * sim * xjv[k];
            float F  = (T >= 0.0f) ? T : 0.01f * T;        // leaky relu
            float ya = 0.9f * xg[k] + 0.1f * F;            // momentum M=0.9
            xg[k] = ya;
            xi[k] = 0.9f * xi[k] + 0.1f * ya;
        }

        if (r == j + 1) {
            // publish updated row j+1 (the xj of step j+1) into the other buffer
            float* nbuf = smem + XJ_OFF + ((j & 1) ? 0 : XJ_STRIDE) + xjb;
            #pragma unroll
            for (int k = 0; k < 16; k += 4)
                *(float4*)(nbuf + k) = make_float4(xi[k], xi[k+1], xi[k+2], xi[k+3]);
        }
    }

    // scatter-back == write to the same source location we gathered from
    float* outy  = out + gbase;
    float* outya = out + (size_t)BB * SS * EE + gbase;
    #pragma unroll
    for (int k = 0; k < 16; k += 4) {
        *(float4*)(outy  + k) = make_float4(xi[k], xi[k+1], xi[k+2], xi[k+3]);
        *(float4*)(outya + k) = make_float4(xg[k], xg[k+1], xg[k+2], xg[k+3]);
    }
}

extern "C" void kernel_launch(void* const* d_in, const int* in_sizes, int n_in,
                              void* d_out, int out_size, void* d_ws, size_t ws_size,
                              hipStream_t stream) {
    const float* x  = (const float*)d_in[0];
    const float* xa = (const float*)d_in[1];
    const float* W  = (const float*)d_in[2];
    float* out = (float*)d_out;
    (void)in_sizes; (void)n_in; (void)out_size; (void)d_ws; (void)ws_size;

    const size_t smem_bytes = (size_t)(2 * STAGE_F + 2 * XJ_STRIDE) * sizeof(float); // ~33 KB
    ncn_fused<<<dim3(BB * NG * NHP), dim3(256), smem_bytes, stream>>>(x, xa, W, out);
}